// EikonalSolverMultiLayer_44916767982053
// MI455X (gfx1250) — compile-verified
//
#include <hip/hip_runtime.h>
#include <cstdint>
#include <cstddef>

// Eikonal multi-layer solver, fused single kernel for MI455X (gfx1250).
// T_{k+1} = min(T_k, conv3x3(T_k, w_k) * (1/sos)), 16 layers, SAME zero pad.
//
// v2: b128 async global<->LDS fill/drain, float4 register-blocked stencil
// (11 DS ops per 4 pixels instead of ~44), 48 KB LDS, 2+ WGs per WGP.

#define NUM_LAYERS 16
#define BATCH      8
#define HEIGHT     1024
#define WIDTH      1024
#define TILE       32
#define HALO       16
#define PAD        64            // TILE + 2*HALO
#define PADSQ      4096          // PAD*PAD
#define NTHREADS   256
#define GROUPS     (PADSQ / 4)   // 1024 float4 groups per tile

__global__ __launch_bounds__(NTHREADS)
void eikonal_fused_kernel(const float* __restrict__ T_init,
                          const float* __restrict__ sos,
                          const float* __restrict__ wts,
                          float* __restrict__ out)
{
    __shared__ __align__(16) float bufA[PADSQ];   // T ping
    __shared__ __align__(16) float bufB[PADSQ];   // T pong
    __shared__ __align__(16) float slo [PADSQ];   // slowness = 1/sos

    const int tid = (int)threadIdx.x;
    const int tx0 = (int)blockIdx.x * TILE - HALO;
    const int ty0 = (int)blockIdx.y * TILE - HALO;
    const size_t img_base = (size_t)blockIdx.z * HEIGHT * WIDTH;

    // ------- tile fill: async global -> LDS, 128-bit per lane -------------
    // Groups are 16B-aligned in LDS and in global (tx0 ≡ 0 mod 4, W % 4 == 0),
    // so a 4-float group is entirely inside or outside the image.
    for (int q = tid; q < GROUPS; q += NTHREADS) {
        const int ly  = q >> 4;            // 64 rows
        const int lxg = (q & 15) << 2;     // 16 groups of 4 per row
        const int p   = ly * PAD + lxg;
        const int gy  = ty0 + ly;
        const int gx  = tx0 + lxg;
        const unsigned la = (unsigned)(uintptr_t)(&bufA[p]);  // LDS byte offset
        const unsigned ls = (unsigned)(uintptr_t)(&slo[p]);
        if ((unsigned)gy < (unsigned)HEIGHT && (unsigned)gx < (unsigned)WIDTH) {
            const size_t g = img_base + (size_t)gy * WIDTH + gx;
            const float* pt = T_init + g;
            const float* ps = sos + g;
            asm volatile("global_load_async_to_lds_b128 %0, %1, off"
                         :: "v"(la), "v"(pt) : "memory");
            asm volatile("global_load_async_to_lds_b128 %0, %1, off"
                         :: "v"(ls), "v"(ps) : "memory");
        } else {
            *(float4*)&bufA[p] = make_float4(0.f, 0.f, 0.f, 0.f); // zero pad
            *(float4*)&slo[p]  = make_float4(1.f, 1.f, 1.f, 1.f); // unused
        }
    }
    asm volatile("s_wait_asynccnt 0" ::: "memory");
    __syncthreads();

    // ------- slowness = 1/sos (identical every layer) ----------------------
    for (int q = tid; q < GROUPS; q += NTHREADS) {
        float4* sp = (float4*)&slo[q << 2];
        float4 s = *sp;
        s.x = 1.0f / s.x; s.y = 1.0f / s.y;
        s.z = 1.0f / s.z; s.w = 1.0f / s.w;
        *sp = s;
    }
    __syncthreads();

    float* cur = bufA;
    float* nxt = bufB;

    // ------- 16 fused stencil layers, float4 register-blocked --------------
    #pragma unroll 1
    for (int layer = 0; layer < NUM_LAYERS; ++layer) {
        const float* wl = wts + layer * 9;   // uniform -> scalar loads
        const float w00 = wl[0], w01 = wl[1], w02 = wl[2];
        const float w10 = wl[3], w11 = wl[4], w12 = wl[5];
        const float w20 = wl[6], w21 = wl[7], w22 = wl[8];

        for (int q = tid; q < GROUPS; q += NTHREADS) {
            const int ly  = q >> 4;
            const int lxg = (q & 15) << 2;
            const int p   = ly * PAD + lxg;
            const float4 c = *(const float4*)&cur[p];   // center row
            float4 o;
            if (ly == 0 || ly == PAD - 1) {
                o = c;                                   // ring row: carry
            } else {
                const float4 a = *(const float4*)&cur[p - PAD]; // row above
                const float4 b = *(const float4*)&cur[p + PAD]; // row below
                // edge scalars (values masked for ring columns)
                const float la0 = cur[p - PAD - 1];
                const float la1 = cur[p - 1];
                const float la2 = cur[p + PAD - 1];
                const float ra0 = cur[p - PAD + 4];
                const float ra1 = cur[p + 4];
                const float ra2 = cur[p + PAD + 4];
                const float4 s  = *(const float4*)&slo[p];

                const int gy = ty0 + ly;
                const int gx = tx0 + lxg;
                const bool oob = ((unsigned)gy >= (unsigned)HEIGHT) ||
                                 ((unsigned)gx >= (unsigned)WIDTH);

                const float c0 = w00*la0 + w01*a.x + w02*a.y
                               + w10*la1 + w11*c.x + w12*c.y
                               + w20*la2 + w21*b.x + w22*b.y;
                const float c1 = w00*a.x + w01*a.y + w02*a.z
                               + w10*c.x + w11*c.y + w12*c.z
                               + w20*b.x + w21*b.y + w22*b.z;
                const float c2 = w00*a.y + w01*a.z + w02*a.w
                               + w10*c.y + w11*c.z + w12*c.w
                               + w20*b.y + w21*b.z + w22*b.w;
                const float c3 = w00*a.z + w01*a.w + w02*ra0
                               + w10*c.z + w11*c.w + w12*ra1
                               + w20*b.z + w21*b.w + w22*ra2;

                o.x = oob ? 0.0f : fminf(c.x, c0 * s.x);
                o.y = oob ? 0.0f : fminf(c.y, c1 * s.y);
                o.z = oob ? 0.0f : fminf(c.z, c2 * s.z);
                o.w = oob ? 0.0f : fminf(c.w, c3 * s.w);
                if (lxg == 0)       o.x = c.x;   // ring column 0
                if (lxg == PAD - 4) o.w = c.w;   // ring column 63
            }
            *(float4*)&nxt[p] = o;
        }
        __syncthreads();
        float* t = cur; cur = nxt; nxt = t;
    }

    // ------- interior write-out: async LDS -> global, 128-bit --------------
    {
        const int q  = tid;                  // 256 threads == 256 groups
        const int iy = q >> 3;               // 32 rows
        const int ix = (q & 7) << 2;         // 8 groups of 4 per row
        const int ly = iy + HALO;
        const int lx = ix + HALO;
        const size_t g = img_base + (size_t)(ty0 + ly) * WIDTH + (tx0 + lx);
        float* pdst = out + g;               // 16B aligned
        const unsigned lsrc = (unsigned)(uintptr_t)(&cur[ly * PAD + lx]);
        asm volatile("global_store_async_from_lds_b128 %0, %1, off"
                     :: "v"(pdst), "v"(lsrc) : "memory");
    }
    // S_ENDPGM implicitly waits for ASYNCcnt == 0.
}

extern "C" void kernel_launch(void* const* d_in, const int* in_sizes, int n_in,
                              void* d_out, int out_size, void* d_ws, size_t ws_size,
                              hipStream_t stream) {
    (void)in_sizes; (void)n_in; (void)out_size; (void)d_ws; (void)ws_size;
    const float* T_init = (const float*)d_in[0];  // [8,1,1024,1024]
    const float* sos    = (const float*)d_in[1];  // [8,1,1024,1024]
    const float* wts    = (const float*)d_in[2];  // [16,1,1,3,3]
    float* out = (float*)d_out;                   // [8,1,1024,1024]

    dim3 grid(WIDTH / TILE, HEIGHT / TILE, BATCH); // 32 x 32 x 8 tiles
    dim3 block(NTHREADS);
    eikonal_fused_kernel<<<grid, block, 0, stream>>>(T_init, sos, wts, out);
}